// SSIMLoss_75557064671871
// MI455X (gfx1250) — compile-verified
//
#include <hip/hip_runtime.h>

// ---------------------------------------------------------------------------
// SSIM loss, fused single pass. f32 WMMA (V_WMMA_F32_16X16X4_F32) implements
// the separable 11x11 Gaussian conv as band-matrix GEMMs in full f32 precision.
// Memory-bound workload (~100MB @ 23.3TB/s => ~4.3us floor); WMMA keeps the
// ~10.6 GFLOP of issued matrix work far off the critical path. Patch staging
// uses CDNA5 async global->LDS copies (ASYNCcnt) when the toolchain exposes
// them, else classic b128 load + ds_store.
// ---------------------------------------------------------------------------

typedef float v2f __attribute__((ext_vector_type(2)));
typedef float v8f __attribute__((ext_vector_type(8)));
typedef int   v4i __attribute__((ext_vector_type(4)));

#define GLOBAL_AS __attribute__((address_space(1)))
#define LDS_AS    __attribute__((address_space(3)))

#if __has_builtin(__builtin_amdgcn_global_load_async_to_lds_b128)
#define HAVE_ASYNC_LDS 1
#else
#define HAVE_ASYNC_LDS 0
#endif

#define IMG_H 512
#define IMG_W 512
#define OUTD  502            // 512 - 11 + 1
#define TILES_X 32           // ceil(502/16)
#define TILES_PER_PLANE (TILES_X * TILES_X)
#define N_PLANES 48          // 16 batches * 3 channels
#define WAVES_PER_BLOCK 8

__constant__ float KVC[11] = {0.001f, 0.0076f, 0.036f, 0.1094f, 0.213f, 0.266f,
                              0.213f, 0.1094f, 0.036f, 0.0076f, 0.001f};

__device__ __forceinline__ void wait_async_copies()
{
#if __has_builtin(__builtin_amdgcn_s_wait_asynccnt)
    __builtin_amdgcn_s_wait_asynccnt(0);
#else
    asm volatile("s_wait_asynccnt 0" ::: "memory");
#endif
    asm volatile("" ::: "memory");   // keep later ds_loads below the copies
}

__global__ __launch_bounds__(256) void ssim_tile_kernel(
    const float* __restrict__ img1, const float* __restrict__ img2,
    float* __restrict__ batch_acc /* 16 floats */)
{
    __shared__ float kwTab[16];
    // per-wave staging (strides 28 / 16 floats are conflict-free for the
    // column-walk access patterns used below on 64 banks)
    __shared__ float P1s[WAVES_PER_BLOCK][26][28];
    __shared__ float P2s[WAVES_PER_BLOCK][26][28];
    __shared__ float Mds[WAVES_PER_BLOCK][28][16];

    const int tid = threadIdx.x;
    if (tid < 16) kwTab[tid] = (tid < 11) ? KVC[tid] : 0.0f;
    __syncthreads();

    const int lane = tid & 31;
    const int wv   = tid >> 5;
    const int wid  = blockIdx.x * WAVES_PER_BLOCK + wv;   // one 16x16 tile per wave
    const int plane = wid / TILES_PER_PLANE;              // 0..47 (n*3+c)
    const int t     = wid - plane * TILES_PER_PLANE;
    const int ty = t >> 5, tx = t & 31;
    const int row0 = ty * 16, col0 = tx * 16;

    const float* __restrict__ p1 = img1 + (size_t)plane * (IMG_H * IMG_W);
    const float* __restrict__ p2 = img2 + (size_t)plane * (IMG_H * IMG_W);

    float (*P1)[28] = P1s[wv];
    float (*P2)[28] = P2s[wv];
    float (*Mid)[16] = Mds[wv];

    // ---- load 26x28 input patch (clamped at image edges) into LDS ----------
    if (row0 + 25 < IMG_H && col0 + 27 < IMG_W) {
        // interior: 26 rows x 7 float4 (16B-aligned both in global and LDS)
#if HAVE_ASYNC_LDS
        for (int i = lane; i < 26 * 7; i += 32) {
            const int r  = i / 7;
            const int c4 = (i - r * 7) * 4;
            const size_t goff = (size_t)(row0 + r) * IMG_W + col0 + c4;
            __builtin_amdgcn_global_load_async_to_lds_b128(
                (GLOBAL_AS v4i*)(p1 + goff), (LDS_AS v4i*)&P1[r][c4], 0, 0);
            __builtin_amdgcn_global_load_async_to_lds_b128(
                (GLOBAL_AS v4i*)(p2 + goff), (LDS_AS v4i*)&P2[r][c4], 0, 0);
        }
        wait_async_copies();
#else
        for (int i = lane; i < 26 * 7; i += 32) {
            const int r  = i / 7;
            const int c4 = (i - r * 7) * 4;
            const float4 a = *(const float4*)(p1 + (size_t)(row0 + r) * IMG_W + col0 + c4);
            const float4 b = *(const float4*)(p2 + (size_t)(row0 + r) * IMG_W + col0 + c4);
            *(float4*)&P1[r][c4] = a;
            *(float4*)&P2[r][c4] = b;
        }
#endif
    } else {
        for (int i = lane; i < 26 * 28; i += 32) {
            const int r = i / 28, c = i - r * 28;
            const int gy = (row0 + r < IMG_H) ? (row0 + r) : (IMG_H - 1);
            const int gx = (col0 + c < IMG_W) ? (col0 + c) : (IMG_W - 1);
            P1[r][c] = p1[(size_t)gy * IMG_W + gx];
            P2[r][c] = p2[(size_t)gy * IMG_W + gx];
        }
    }
    // zero Mid pad rows 26..27 (their weights are 0, but 0*NaN => NaN in WMMA)
    Mid[26 + (lane >> 4)][lane & 15] = 0.0f;

    // ---- band-matrix weights, per WMMA lane layout --------------------------
    // For both the horizontal-B and vertical-A operands the per-lane value is
    // g[K - (lane&15)] with K = 4q + (lane>>4)*2 + v  (v = VGPR 0/1).
    const int nM   = lane & 15;
    const int koff = (lane >> 4) * 2;
    v2f Wq[7];
#pragma unroll
    for (int q = 0; q < 7; ++q) {
        const int k0 = 4 * q + koff;
        const int d0 = k0 - nM, d1 = k0 + 1 - nM;
        Wq[q].x = kwTab[((unsigned)d0 <= 10u) ? d0 : 11];
        Wq[q].y = kwTab[((unsigned)d1 <= 10u) ? d1 : 11];
    }

    // ---- 5 fields: x, y, x*x, y*y, x*y -------------------------------------
    v8f outF[5];
#pragma unroll
    for (int f = 0; f < 5; ++f) {
        // horizontal pass: Mid rows 0..15 (s=0) and 10..25 (s=1)
#pragma unroll
        for (int s = 0; s < 2; ++s) {
            const int rowbase = s * 10;
            const int prow = rowbase + nM;   // A-operand: M = lane&15
            v8f acc = {0.f, 0.f, 0.f, 0.f, 0.f, 0.f, 0.f, 0.f};
#pragma unroll
            for (int q = 0; q < 7; ++q) {
                const int c0 = 4 * q + koff;
                const float a0 = P1[prow][c0], a1 = P1[prow][c0 + 1];
                const float b0 = P2[prow][c0], b1 = P2[prow][c0 + 1];
                v2f A;
                switch (f) {
                    case 0:  A.x = a0;      A.y = a1;      break;
                    case 1:  A.x = b0;      A.y = b1;      break;
                    case 2:  A.x = a0 * a0; A.y = a1 * a1; break;
                    case 3:  A.x = b0 * b0; A.y = b1 * b1; break;
                    default: A.x = a0 * b0; A.y = a1 * b1; break;
                }
                acc = __builtin_amdgcn_wmma_f32_16x16x4_f32(
                    false, A, false, Wq[q], (short)0, acc, false, false);
            }
            // scatter D (VGPR r -> rows rowbase+r / rowbase+r+8) into Mid
#pragma unroll
            for (int r = 0; r < 8; ++r) {
                const int mr = rowbase + r + ((lane >> 4) << 3);
                Mid[mr][nM] = acc[r];
            }
        }
        // vertical pass: Out = Wv(16x28) x Mid(28x16)
        v8f o = {0.f, 0.f, 0.f, 0.f, 0.f, 0.f, 0.f, 0.f};
#pragma unroll
        for (int q = 0; q < 7; ++q) {
            const int k0 = 4 * q + koff;
            v2f B;
            B.x = Mid[k0][nM];
            B.y = Mid[k0 + 1][nM];
            o = __builtin_amdgcn_wmma_f32_16x16x4_f32(
                false, Wq[q], false, B, (short)0, o, false, false);
        }
        outF[f] = o;
    }

    // ---- SSIM map + masked reduction ---------------------------------------
    const float C1 = 6.5025f, C2 = 58.5225f;
    float sum = 0.0f;
#pragma unroll
    for (int r = 0; r < 8; ++r) {
        const int m  = r + ((lane >> 4) << 3);
        const int oy = row0 + m, ox = col0 + nM;
        const float mu1 = outF[0][r], mu2 = outF[1][r];
        const float e11 = outF[2][r], e22 = outF[3][r], e12 = outF[4][r];
        const float mu1s = mu1 * mu1, mu2s = mu2 * mu2, mu12 = mu1 * mu2;
        const float s1 = e11 - mu1s, s2 = e22 - mu2s, s12 = e12 - mu12;
        const float cs   = (2.0f * s12 + C2) / (s1 + s2 + C2);
        const float ssim = (2.0f * mu12 + C1) / (mu1s + mu2s + C1) * cs;
        sum += (oy < OUTD && ox < OUTD) ? ssim : 0.0f;   // cndmask, no branch
    }
#pragma unroll
    for (int off = 16; off > 0; off >>= 1)
        sum += __shfl_xor(sum, off, 32);
    if (lane == 0)
        atomicAdd(&batch_acc[plane / 3], sum);
}

__global__ void ssim_finalize_kernel(const float* __restrict__ acc,
                                     float* __restrict__ out)
{
    const int i = threadIdx.x;
    if (i < 16)
        out[i] = 1.0f - acc[i] * (1.0f / (3.0f * 502.0f * 502.0f));
}

extern "C" void kernel_launch(void* const* d_in, const int* in_sizes, int n_in,
                              void* d_out, int out_size, void* d_ws, size_t ws_size,
                              hipStream_t stream)
{
    const float* img1 = (const float*)d_in[0];
    const float* img2 = (const float*)d_in[1];
    float* out = (float*)d_out;
    float* acc = (float*)d_ws;

    hipMemsetAsync(acc, 0, 16 * sizeof(float), stream);

    const int total_waves  = N_PLANES * TILES_PER_PLANE;     // 49152
    const int blocks       = total_waves / WAVES_PER_BLOCK;  // 6144
    ssim_tile_kernel<<<blocks, 32 * WAVES_PER_BLOCK, 0, stream>>>(img1, img2, acc);
    ssim_finalize_kernel<<<1, 16, 0, stream>>>(acc, out);
}